// BpmllLoss_20693152432248
// MI455X (gfx1250) — compile-verified
//
#include <hip/hip_runtime.h>

typedef __attribute__((ext_vector_type(2))) float v2f;
typedef __attribute__((ext_vector_type(8))) float v8f;

#define C_DIM 256

// One wave (32 lanes) per block; each wave reduces a 16-row x 256-col tile.
// Row sums are computed as WMMA chains: D = A * Ones + C  (V_WMMA_F32_16X16X4_F32).
//
// A-matrix 16x4 f32 layout (ISA 7.12.2):
//   lanes 0-15 : M = lane,     VGPR0 = K0, VGPR1 = K1
//   lanes 16-31: M = lane-16,  VGPR0 = K2, VGPR1 = K3
// C/D 16x16 f32 layout: lane 0 holds D[0..7][0] in VGPR0..7, lane 16 holds D[8..15][0].
__global__ __launch_bounds__(32)
void bpmll_row_factors(const float* __restrict__ x,
                       const float* __restrict__ t,
                       float* __restrict__ row_loss)
{
    const int lane = threadIdx.x;           // 0..31, wave32
    const int row0 = blockIdx.x * 16;       // 16 rows per wave
    const int m    = lane & 15;             // row within tile
    const int ksel = (lane >> 4) << 1;      // 0 for lanes 0-15, 2 for lanes 16-31

    const float* xrow = x + (size_t)(row0 + m) * C_DIM;
    const float* trow = t + (size_t)(row0 + m) * C_DIM;

    v2f bones; bones.x = 1.0f; bones.y = 1.0f;   // every element of B is 1.0
    v8f accU = {};   // sum_k exp(-x)*t
    v8f accV = {};   // sum_k exp(+x)*(1-t)
    v8f accD = {};   // sum_k t  (= dim)

    for (int k0 = 0; k0 < C_DIM; k0 += 4) {
        float x0 = xrow[k0 + ksel + 0];
        float x1 = xrow[k0 + ksel + 1];
        float t0 = trow[k0 + ksel + 0];
        float t1 = trow[k0 + ksel + 1];

        float en0 = __expf(-x0), en1 = __expf(-x1);
        float ep0 = __expf(x0),  ep1 = __expf(x1);

        v2f aU; aU.x = en0 * t0;          aU.y = en1 * t1;
        v2f aV; aV.x = ep0 * (1.0f - t0); aV.y = ep1 * (1.0f - t1);
        v2f aD; aD.x = t0;                aD.y = t1;

        accU = __builtin_amdgcn_wmma_f32_16x16x4_f32(
            false, aU, false, bones, (short)0, accU, false, false);
        accV = __builtin_amdgcn_wmma_f32_16x16x4_f32(
            false, aV, false, bones, (short)0, accV, false, false);
        accD = __builtin_amdgcn_wmma_f32_16x16x4_f32(
            false, aD, false, bones, (short)0, accD, false, false);
    }

    // Lane 0 owns rows 0..7 (N=0 column), lane 16 owns rows 8..15.
    if ((lane & 15) == 0) {
        const int mbase = (lane >> 4) * 8;
        #pragma unroll
        for (int i = 0; i < 8; ++i) {
            float u = accU[i];
            float v = accV[i];
            float d = accD[i];
            float norm = d * ((float)C_DIM - d);
            row_loss[row0 + mbase + i] = (u * v) / norm;
        }
    }
}

// Deterministic fixed-order tree reduction of the per-row losses.
__global__ __launch_bounds__(256)
void bpmll_reduce(const float* __restrict__ row_loss, float* __restrict__ out, int n)
{
    __shared__ float sm[256];
    const int tid = threadIdx.x;
    float s = 0.0f;
    for (int i = tid; i < n; i += 256) s += row_loss[i];
    sm[tid] = s;
    __syncthreads();
    for (int off = 128; off > 0; off >>= 1) {
        if (tid < off) sm[tid] += sm[tid + off];
        __syncthreads();
    }
    if (tid == 0) out[0] = sm[0];
}

extern "C" void kernel_launch(void* const* d_in, const int* in_sizes, int n_in,
                              void* d_out, int out_size, void* d_ws, size_t ws_size,
                              hipStream_t stream)
{
    const float* x = (const float*)d_in[0];   // [B, C] f32
    const float* t = (const float*)d_in[1];   // [B, C] f32 in {0,1}
    const int B = in_sizes[0] / C_DIM;        // 2048

    float* row_loss = (float*)d_ws;           // B * 4 bytes of scratch

    bpmll_row_factors<<<B / 16, 32, 0, stream>>>(x, t, row_loss);
    bpmll_reduce<<<1, 256, 0, stream>>>(row_loss, (float*)d_out, B);
}